// Canny_68899865362473
// MI455X (gfx1250) — compile-verified
//
#include <hip/hip_runtime.h>
#include <stdint.h>

typedef __attribute__((ext_vector_type(2))) float v2f;
typedef __attribute__((ext_vector_type(8))) float v8f;

#define USE_ASYNC_LDS 1

// Gaussian 5-tap weights (sigma=1, normalized), g[j] for j in [0,4]
__device__ __forceinline__ float gw(int j) {
  float r = 0.0f;
  r = (j == 0 || j == 4) ? 0.054488685f : r;
  r = (j == 1 || j == 3) ? 0.244201342f : r;
  r = (j == 2)           ? 0.402619947f : r;
  return r;
}

// --------------------------------------------------------------------------
// Kernel A: separable Gaussian blur. One wave handles one 16x16 output tile
// of one (batch*channel) plane. Horizontal pass is scalar from LDS; vertical
// pass is a banded-matrix multiply on the matrix pipe:
//   Blur(16x16) = Gv(16x20) x Hb(20x16), five V_WMMA_F32_16X16X4_F32 (K=4).
// The constant band matrix Gv[m][k] = g[k-m] lives in an LDS table so the
// A-operand is two conflict-free ds_loads per K-chunk (no cndmask chains
// between WMMAs).
// --------------------------------------------------------------------------
__global__ __launch_bounds__(256) void blur16_wmma(const float* __restrict__ img,
                                                   float* __restrict__ blurred) {
  const int lane = threadIdx.x & 31;
  const int wv   = threadIdx.x >> 5;
  const int tileId = blockIdx.x * 8 + wv;       // 96 planes * 1024 tiles
  const int plane  = tileId >> 10;
  const int t      = tileId & 1023;
  const int gy0 = (t >> 5) << 4;
  const int gx0 = (t & 31) << 4;
  const float* src = img     + (size_t)plane * (512 * 512);
  float*       dst = blurred + (size_t)plane * (512 * 512);

  __shared__ float lds[8][20 * 20 + 20 * 16];
  __shared__ float sGv[16 * 20];  // band matrix, shared by all 8 waves
  float* E  = &lds[wv][0];        // 20x20 input tile (halo 2), row stride 20
  float* Hb = &lds[wv][400];      // 20x16 horizontally blurred, row stride 16

  // stage 1a: band-matrix table (workgroup-wide constant)
  for (int i = threadIdx.x; i < 320; i += 256) {
    int mm = i / 20, kk = i - mm * 20;
    sGv[i] = gw(kk - mm);
  }
  // stage 1b: zero-padded ('SAME') halo load
  for (int i = lane; i < 400; i += 32) {
    int r = i / 20, c = i - r * 20;
    int gr = gy0 + r - 2, gc = gx0 + c - 2;
    float v = 0.0f;
    if ((unsigned)gr < 512u && (unsigned)gc < 512u) v = src[gr * 512 + gc];
    E[i] = v;
  }
  __syncthreads();

  // stage 2: horizontal 5-tap (symmetric) blur, rows 0..19 x cols 0..15
  for (int i = lane; i < 320; i += 32) {
    int r = i >> 4, c = i & 15;
    const float* e = E + r * 20 + c;
    Hb[i] = 0.054488685f * (e[0] + e[4]) + 0.244201342f * (e[1] + e[3])
          + 0.402619947f * e[2];
  }
  __syncthreads();

  // stage 3: vertical blur as banded GEMM on the WMMA pipe.
  // A layout (16x4 f32): lane l: M = l%16, K = 2*(l>>4) + component.
  // B layout (4x16 f32): lane l: N = l%16, K = 2*(l>>4) + component.
  // D layout (16x16 f32, 8 VGPRs): vgpr v: M = v + 8*(l>>4), N = l%16.
  const int m    = lane & 15;   // A's M and B's N
  const int half = lane >> 4;
  const float* arow = sGv + m * 20;
  v8f acc = {};
#pragma unroll
  for (int q = 0; q < 5; ++q) {
    const int k0 = 4 * q + 2 * half;
    v2f a, b;
    a.x = arow[k0];             // Gv[m][k] = g[k-m] on the band, else 0
    a.y = arow[k0 + 1];
    b.x = Hb[k0 * 16 + m];
    b.y = Hb[(k0 + 1) * 16 + m];
    acc = __builtin_amdgcn_wmma_f32_16x16x4_f32(false, a, false, b,
                                                (short)0, acc, false, false);
  }

#pragma unroll
  for (int v = 0; v < 8; ++v) {
    dst[(size_t)(gy0 + v + 8 * half) * 512 + gx0 + m] = acc[v];
  }
}

// --------------------------------------------------------------------------
// Async global -> LDS staging (CDNA5). Per-lane copy; ASYNCcnt tracked.
// --------------------------------------------------------------------------
__device__ __forceinline__ void asyncLoadB32(uint32_t ldsOff, const float* g) {
  asm volatile("global_load_async_to_lds_b32 %0, %1, off"
               :: "v"(ldsOff), "v"(g) : "memory");
}
__device__ __forceinline__ void waitAsync0() {
#if __has_builtin(__builtin_amdgcn_s_wait_asynccnt)
  __builtin_amdgcn_s_wait_asynccnt(0);
#else
  asm volatile("s_wait_asynccnt 0" ::: "memory");
#endif
}

// --------------------------------------------------------------------------
// Kernel B: fused Sobel + magnitude + orientation + 8-dir NMS + thresholds.
// 256 threads per 32x32 output tile; blurred halo(2) tile staged via async
// copies. grad_mag kept on a 34x34 LDS grid (zero outside image, matching the
// zero-padded directional conv of the reference).
// --------------------------------------------------------------------------
__global__ __launch_bounds__(256) void canny_stage2(const float* __restrict__ blurred,
                                                    float* __restrict__ gm_out,
                                                    float* __restrict__ ori_out,
                                                    float* __restrict__ thin_out,
                                                    float* __restrict__ thr_out,
                                                    float* __restrict__ early_out) {
  __shared__ float sblur[3][36 * 36];
  __shared__ float smag[34 * 34];
  __shared__ float sgx[34 * 34];
  __shared__ float sgy[34 * 34];

  const int tid = threadIdx.x;
  const int b   = blockIdx.x >> 8;       // 256 tiles per batch image
  const int t   = blockIdx.x & 255;
  const int gy0 = (t >> 4) << 5;
  const int gx0 = (t & 15) << 5;
  const float* bp = blurred + (size_t)b * 3 * 262144;

  // stage 1: async-stage blurred 36x36 halo tile, 3 channels, zero-padded
  for (int c = 0; c < 3; ++c) {
    const float* src = bp + c * 262144;
    for (int i = tid; i < 36 * 36; i += 256) {
      int r = i / 36, cc = i - r * 36;
      int gr = gy0 + r - 2, gc = gx0 + cc - 2;
      if ((unsigned)gr < 512u && (unsigned)gc < 512u) {
#if USE_ASYNC_LDS
        asyncLoadB32((uint32_t)(uintptr_t)&sblur[c][i], src + gr * 512 + gc);
#else
        sblur[c][i] = src[gr * 512 + gc];
#endif
      } else {
        sblur[c][i] = 0.0f;
      }
    }
  }
#if USE_ASYNC_LDS
  waitAsync0();
#endif
  __syncthreads();

  // stage 2: Sobel per channel -> grad_mag / gx_sum / gy_sum on 34x34 grid
  for (int i = tid; i < 34 * 34; i += 256) {
    int r = i / 34, cc = i - r * 34;
    int gr = gy0 + r - 1, gc = gx0 + cc - 1;
    float mag = 0.0f, gxs = 0.0f, gys = 0.0f;
    if ((unsigned)gr < 512u && (unsigned)gc < 512u) {
#pragma unroll
      for (int c = 0; c < 3; ++c) {
        const float* p = &sblur[c][(r + 1) * 36 + (cc + 1)];
        float p00 = p[-37], p01 = p[-36], p02 = p[-35];
        float p10 = p[-1],                p12 = p[1];
        float p20 = p[35],  p21 = p[36],  p22 = p[37];
        float gx = (p00 - p02) + 2.0f * (p10 - p12) + (p20 - p22);
        float gy = (p00 + 2.0f * p01 + p02) - (p20 + 2.0f * p21 + p22);
        mag += sqrtf(gx * gx + gy * gy);
        gxs += gx;
        gys += gy;
      }
    }
    smag[i] = mag;
    sgx[i]  = gxs;
    sgy[i]  = gys;
  }
  __syncthreads();

  // stage 3: orientation, NMS along quantized direction, thresholds
  for (int i = tid; i < 1024; i += 256) {
    int r = i >> 5, cc = i & 31;
    int mi = (r + 1) * 34 + (cc + 1);
    float mag = smag[mi];
    float o   = atan2f(sgy[mi], sgx[mi]) * 57.29577951308232f + 180.0f;
    float oq  = rintf(o * (1.0f / 45.0f));       // RNE, matches jnp.round
    int idx   = ((int)oq) & 7;                    // pos_idx; neg = idx+4 mod 8
    // LDS offset of direction d (stride 34): {+1,+35,+34,+33,-1,-35,-34,-33}
    int i3  = idx & 3;
    int o03 = (i3 == 0) ? 1 : (36 - i3);
    int off = (idx & 4) ? -o03 : o03;             // off[idx+4] == -off[idx]
    float pf = mag - smag[mi + off];
    float nf = mag - smag[mi - off];
    float thin = (fminf(pf, nf) > 0.0f) ? mag : 0.0f;

    size_t oo = (size_t)b * 262144 + (size_t)(gy0 + r) * 512 + (gx0 + cc);
    gm_out[oo]    = mag;
    ori_out[oo]   = oq * 45.0f;
    thin_out[oo]  = thin;
    thr_out[oo]   = (thin < 2.0f) ? 0.0f : thin;
    early_out[oo] = (mag < 2.0f) ? 0.0f : mag;
  }
}

extern "C" void kernel_launch(void* const* d_in, const int* in_sizes, int n_in,
                              void* d_out, int out_size, void* d_ws, size_t ws_size,
                              hipStream_t stream) {
  (void)in_sizes; (void)n_in; (void)out_size; (void)d_ws; (void)ws_size;
  const float* img = (const float*)d_in[0];
  float* out = (float*)d_out;
  // Output tuple, concatenated flat:
  float* blurred = out;                    // 32*3*512*512
  float* gm      = out + 25165824u;        // 32*1*512*512
  float* ori     = out + 33554432u;
  float* thin    = out + 41943040u;
  float* thr     = out + 50331648u;
  float* early   = out + 58720256u;

  // Kernel A: 96 planes * (32*32) tiles, 8 waves (tiles) per 256-thread block
  blur16_wmma<<<dim3(12288), dim3(256), 0, stream>>>(img, blurred);
  // Kernel B: 32 batches * 256 tiles of 32x32
  canny_stage2<<<dim3(8192), dim3(256), 0, stream>>>(blurred, gm, ori, thin, thr, early);
}